// EvolveGCNH_46033459478916
// MI455X (gfx1250) — compile-verified
//
#include <hip/hip_runtime.h>
#include <math.h>
#include <float.h>

#define T_STEPS 4
#define NNODES  50000
#define NEDGES  800000
#define FDIM    256
#define HDIM    128
#define CHID    512
#define COUT    2
#define SLOPE   (11.0f/48.0f)

typedef float v2f __attribute__((ext_vector_type(2)));
typedef float v8f __attribute__((ext_vector_type(8)));
typedef int   v4i __attribute__((ext_vector_type(4)));
typedef __attribute__((address_space(1))) v4i* gptr_v4i;   // global int4*
typedef __attribute__((address_space(3))) v4i* lptr_v4i;   // LDS int4*

// Optional gfx1250 async global->LDS DMA path (ASYNCcnt-tracked)
#if defined(__has_builtin)
#if __has_builtin(__builtin_amdgcn_global_load_async_to_lds_b128) && \
    __has_builtin(__builtin_amdgcn_s_wait_asynccnt)
#define HAVE_ASYNC_LDS 1
#endif
#endif
#ifndef HAVE_ASYNC_LDS
#define HAVE_ASYNC_LDS 0
#endif

// CDNA5 fp32 WMMA: D(16x16 f32) = A(16x4 f32) * B(4x16 f32) + C
__device__ __forceinline__ v8f wmma4(v2f a, v2f b, v8f c) {
    return __builtin_amdgcn_wmma_f32_16x16x4_f32(false, a, false, b, (short)0, c,
                                                 false, false);
}

// ---------------------------------------------------------------- utilities
__global__ void k_fill(float* p, float v, int n) {
    int i = blockIdx.x * blockDim.x + threadIdx.x;
    if (i < n) p[i] = v;
}

__global__ void k_invnorm(const float* sc, int d, float* out) {
    __shared__ float red[256];
    float s = 0.f;
    for (int i = threadIdx.x; i < d; i += 256) { float v = sc[i]; s += v * v; }
    red[threadIdx.x] = s;
    __syncthreads();
    for (int o = 128; o > 0; o >>= 1) {
        if (threadIdx.x < o) red[threadIdx.x] += red[threadIdx.x + o];
        __syncthreads();
    }
    if (threadIdx.x == 0) out[0] = 1.0f / fmaxf(sqrtf(red[0]), 1e-6f);
}

// wave-per-row dot: scores[n] = dot(x[n,:], scorer) * invnorm
__global__ void k_scores(const float* __restrict__ x, const float* __restrict__ sc,
                         const float* invn, float* scores, int d) {
    int w = (blockIdx.x * blockDim.x + threadIdx.x) >> 5;
    int lane = threadIdx.x & 31;
    if (w >= NNODES) return;
    const float* row = x + (size_t)w * d;
    float s = 0.f;
    for (int i = lane; i < d; i += 32) s += row[i] * sc[i];
    for (int o = 16; o > 0; o >>= 1) s += __shfl_down(s, o, 32);
    if (lane == 0) scores[w] = s * invn[0];
}

// top-128 of 50000 scores: whole score array resident in LDS (200KB < 320KB/WGP)
__global__ __launch_bounds__(1024) void k_topk(const float* __restrict__ scores,
                                               int* idx_out, float* val_out) {
    extern __shared__ float smem[];
    float* sv = smem;                      // NNODES
    float* rv = smem + NNODES;             // 1024
    int*   ri = (int*)(rv + 1024);         // 1024
    int tid = threadIdx.x;
    for (int i = tid; i < NNODES; i += 1024) sv[i] = scores[i];
    __syncthreads();
    for (int j = 0; j < HDIM; ++j) {
        float best = -FLT_MAX; int bi = 0x7fffffff;
        for (int i = tid; i < NNODES; i += 1024) {
            float v = sv[i];
            if (v > best || (v == best && i < bi)) { best = v; bi = i; }
        }
        rv[tid] = best; ri[tid] = bi;
        __syncthreads();
        for (int o = 512; o > 0; o >>= 1) {
            if (tid < o) {
                float v2 = rv[tid + o]; int i2 = ri[tid + o];
                if (v2 > rv[tid] || (v2 == rv[tid] && i2 < ri[tid])) {
                    rv[tid] = v2; ri[tid] = i2;
                }
            }
            __syncthreads();
        }
        if (tid == 0) {
            int m = ri[0];
            idx_out[j] = m; val_out[j] = rv[0];
            sv[m] = -FLT_MAX;
        }
        __syncthreads();
    }
}

// X_tilde[r, j] = x[idx[j], r] * tanh(val[j]);   X_tilde is [d, 128]
__global__ void k_xtilde(const float* __restrict__ x, const int* __restrict__ idx,
                         const float* __restrict__ val, float* Xt, int d) {
    int gid = blockIdx.x * blockDim.x + threadIdx.x;
    if (gid >= d * HDIM) return;
    int r = gid >> 7, j = gid & 127;
    Xt[gid] = x[(size_t)idx[j] * d + r] * tanhf(val[j]);
}

// GRU gate: G = act(A1 @ B1 + A2 @ B2 + bias)  with [d,d]@[d,128]; WMMA f32 tiles.
// mode 0: sigmoid -> out.   mode 1: tanh + combine: Wout = (1-u)*Q + u*hcap
__global__ __launch_bounds__(256) void k_gru_gate(
        const float* __restrict__ A1, const float* __restrict__ B1,
        const float* __restrict__ A2, const float* __restrict__ B2,
        const float* __restrict__ bias, int d, int mode,
        float* out, const float* u, const float* Q, float* Wout) {
    int lane = threadIdx.x & 31, wid = threadIdx.x >> 5;
    int colTiles = HDIM / 16;
    int ntiles = (d / 16) * colTiles;
    int tile = blockIdx.x * 8 + wid;
    if (tile >= ntiles) return;
    int m0 = (tile / colTiles) * 16, n0 = (tile % colTiles) * 16;
    int laneM = lane & 15, hi = lane >> 4;
    v8f acc = {};
    const float* A1row = A1 + (size_t)(m0 + laneM) * d;
    for (int k0 = 0; k0 < d; k0 += 4) {
        int kk = k0 + hi * 2;
        v2f a = *(const v2f*)(A1row + kk);
        v2f b;
        b.x = B1[(size_t)kk * HDIM + n0 + laneM];
        b.y = B1[(size_t)(kk + 1) * HDIM + n0 + laneM];
        acc = wmma4(a, b, acc);
    }
    const float* A2row = A2 + (size_t)(m0 + laneM) * d;
    for (int k0 = 0; k0 < d; k0 += 4) {
        int kk = k0 + hi * 2;
        v2f a = *(const v2f*)(A2row + kk);
        v2f b;
        b.x = B2[(size_t)kk * HDIM + n0 + laneM];
        b.y = B2[(size_t)(kk + 1) * HDIM + n0 + laneM];
        acc = wmma4(a, b, acc);
    }
    int n = n0 + laneM;
#pragma unroll
    for (int vi = 0; vi < 8; ++vi) {
        int m = m0 + vi + hi * 8;
        float v = acc[vi] + bias[(size_t)m * HDIM + n];
        if (mode == 0) {
            out[(size_t)m * HDIM + n] = 1.0f / (1.0f + expf(-v));
        } else {
            float hc = tanhf(v);
            float uu = u[(size_t)m * HDIM + n];
            float qq = Q[(size_t)m * HDIM + n];
            Wout[(size_t)m * HDIM + n] = (1.0f - uu) * qq + uu * hc;
        }
    }
}

__global__ void k_had(const float* a, const float* b, float* c, int n) {
    int i = blockIdx.x * blockDim.x + threadIdx.x;
    if (i < n) c[i] = a[i] * b[i];
}

// Big WMMA matmul: C[M,Nc] = act((rowScale.*A)[M,K] @ B[K,Nc] + bias)
// Block = 8 waves owning one 16-row strip; A strip staged once into LDS
// (async DMA when available), 8 column tiles per block pass.
// grid = (M/16, Nc/128). Requires K in {128, 256}, Nc multiple of 128.
#define LDSPAD 4
__global__ __launch_bounds__(256) void k_mm(
        const float* __restrict__ A, const float* __restrict__ B, float* C,
        int M, int K, int Nc, const float* rowScale, const float* bias, int relu) {
    __shared__ float ldsA[16 * (256 + LDSPAD)];
    int tid = threadIdx.x;
    int lane = tid & 31, wid = tid >> 5;
    int m0 = blockIdx.x * 16;
    int n0 = (blockIdx.y * 8 + wid) * 16;
    if (m0 >= M) return;
    int stride = K + LDSPAD;
    const float* Abase = A + (size_t)m0 * K;
    int total4 = (16 * K) >> 2;
#if HAVE_ASYNC_LDS
    for (int f4 = tid; f4 < total4; f4 += 256) {
        int f = f4 * 4;
        int r = f / K, k = f - r * K;
        __builtin_amdgcn_global_load_async_to_lds_b128(
            (gptr_v4i)(Abase + f), (lptr_v4i)(&ldsA[r * stride + k]), 0, 0);
    }
    __builtin_amdgcn_s_wait_asynccnt(0);
    __syncthreads();
#else
    for (int f4 = tid; f4 < total4; f4 += 256) {
        int f = f4 * 4;
        int r = f / K, k = f - r * K;
        float4 v = *(const float4*)(Abase + f);
        *(float4*)(&ldsA[r * stride + k]) = v;
    }
    __syncthreads();
#endif
    int laneM = lane & 15, hi = lane >> 4;
    float sc = rowScale ? rowScale[m0 + laneM] : 1.0f;
    const float* ldsRow = &ldsA[laneM * stride];
    const float* Bcol = B + n0 + laneM;
    v8f acc = {};
    for (int k0 = 0; k0 < K; k0 += 4) {
        int kk = k0 + hi * 2;
        v2f a = *(const v2f*)(ldsRow + kk);
        a.x *= sc; a.y *= sc;
        v2f b;
        b.x = Bcol[(size_t)kk * Nc];
        b.y = Bcol[(size_t)(kk + 1) * Nc];
        acc = wmma4(a, b, acc);
    }
    int n = n0 + laneM;
#pragma unroll
    for (int vi = 0; vi < 8; ++vi) {
        int m = m0 + vi + hi * 8;
        float v = acc[vi];
        if (bias) v += bias[n];
        if (relu) v = fmaxf(v, 0.0f);
        C[(size_t)m * Nc + n] = v;
    }
}

__global__ void k_deg(const int* __restrict__ src, const int* __restrict__ dst,
                      float* dout, float* din) {
    int e = blockIdx.x * blockDim.x + threadIdx.x;
    if (e >= NEDGES) return;
    atomicAdd(&dout[src[e]], 1.0f);
    atomicAdd(&din[dst[e]], 1.0f);
}

__global__ void k_degfin(float* s, int n) {
    int i = blockIdx.x * blockDim.x + threadIdx.x;
    if (i < n) s[i] = rsqrtf(fmaxf(s[i], 1.0f));
}

// agg[dst[e], :] += h[src[e], :]  — thread per (edge, 4 channels):
// b128 gather + 4 f32 atomics; wave covers one full 128-ch edge row.
__global__ void k_scatter(const int* __restrict__ src, const int* __restrict__ dst,
                          const float* __restrict__ h, float* agg) {
    long gid = (long)blockIdx.x * blockDim.x + threadIdx.x;
    if (gid >= (long)NEDGES * (HDIM / 4)) return;
    int e = (int)(gid >> 5);
    int q = (int)(gid & 31) * 4;
    int s = src[e], d = dst[e];
    // prefetch a future edge's source row behind the atomic stream
    int e2 = e + 2048;
    if (q == 0 && e2 < NEDGES)
        __builtin_prefetch(h + (size_t)src[e2] * HDIM, 0, 1);
    float4 v = *(const float4*)(h + (size_t)s * HDIM + q);
    float* ap = agg + (size_t)d * HDIM + q;
    atomicAdd(ap + 0, v.x);
    atomicAdd(ap + 1, v.y);
    atomicAdd(ap + 2, v.z);
    atomicAdd(ap + 3, v.w);
}

__global__ void k_finalize(const float* __restrict__ agg, const float* __restrict__ sin,
                           float* out, int n) {
    int i = blockIdx.x * blockDim.x + threadIdx.x;
    if (i >= n) return;
    float v = agg[i] * sin[i >> 7];
    out[i] = v >= 0.f ? v : SLOPE * v;
}

// out[n, 0:2] = hidden[n,:] @ W2 + b2  — wave per row, dual dot
__global__ void k_mlp2(const float* __restrict__ hid, const float* __restrict__ W2,
                       const float* __restrict__ b2, float* out) {
    int w = (blockIdx.x * blockDim.x + threadIdx.x) >> 5;
    int lane = threadIdx.x & 31;
    if (w >= NNODES) return;
    const float* row = hid + (size_t)w * CHID;
    float p0 = 0.f, p1 = 0.f;
    for (int i = lane; i < CHID; i += 32) {
        float v = row[i];
        p0 += v * W2[i * 2];
        p1 += v * W2[i * 2 + 1];
    }
    for (int o = 16; o > 0; o >>= 1) {
        p0 += __shfl_down(p0, o, 32);
        p1 += __shfl_down(p1, o, 32);
    }
    if (lane == 0) { out[w * 2] = p0 + b2[0]; out[w * 2 + 1] = p1 + b2[1]; }
}

extern "C" void kernel_launch(void* const* d_in, const int* in_sizes, int n_in,
                              void* d_out, int out_size, void* d_ws, size_t ws_size,
                              hipStream_t stream) {
    (void)in_sizes; (void)n_in; (void)out_size; (void)ws_size;
    const float* feats = (const float*)d_in[0];
    const int*   edges = (const int*)d_in[1];
    const float* scorer[2] = {(const float*)d_in[2], (const float*)d_in[3]};
    const float* gcnW[2]   = {(const float*)d_in[4], (const float*)d_in[5]};
    const float* gw[2][9];
    for (int l = 0; l < 2; ++l)
        for (int i = 0; i < 9; ++i) gw[l][i] = (const float*)d_in[6 + l * 9 + i];
    const float* mW1 = (const float*)d_in[24];
    const float* mb1 = (const float*)d_in[25];
    const float* mW2 = (const float*)d_in[26];
    const float* mb2 = (const float*)d_in[27];

    char* ws = (char*)d_ws;
    size_t off = 0;
    auto alloc = [&](size_t bytes) -> void* {
        void* p = ws + off;
        off = (off + bytes + 255) & ~(size_t)255;
        return p;
    };
    float* featsA = (float*)alloc((size_t)T_STEPS * NNODES * HDIM * 4);
    float* hidden = (float*)alloc((size_t)NNODES * CHID * 4);
    float* hbuf   = (float*)alloc((size_t)NNODES * HDIM * 4);
    float* agg    = (float*)alloc((size_t)NNODES * HDIM * 4);
    float* scores = (float*)alloc((size_t)NNODES * 4);
    float* so     = (float*)alloc((size_t)NNODES * 4);
    float* si     = (float*)alloc((size_t)NNODES * 4);
    int*   tidx   = (int*)alloc(HDIM * 4);
    float* tval   = (float*)alloc(HDIM * 4);
    float* invn   = (float*)alloc(256);
    float* Xt     = (float*)alloc((size_t)FDIM * HDIM * 4);
    float* Wq     = (float*)alloc((size_t)FDIM * HDIM * 4);
    float* Wn     = (float*)alloc((size_t)FDIM * HDIM * 4);
    float* ubuf   = (float*)alloc((size_t)FDIM * HDIM * 4);
    float* rbuf   = (float*)alloc((size_t)FDIM * HDIM * 4);
    float* rqb    = (float*)alloc((size_t)FDIM * HDIM * 4);

    for (int layer = 0; layer < 2; ++layer) {
        int d = (layer == 0) ? FDIM : HDIM;
        (void)hipMemcpyAsync(Wq, gcnW[layer], (size_t)d * HDIM * 4,
                             hipMemcpyDeviceToDevice, stream);
        for (int t = 0; t < T_STEPS; ++t) {
            const float* x = (layer == 0) ? feats + (size_t)t * NNODES * FDIM
                                          : featsA + (size_t)t * NNODES * HDIM;
            // ---- top-k pooling ----
            k_invnorm<<<1, 256, 0, stream>>>(scorer[layer], d, invn);
            k_scores<<<(NNODES + 7) / 8, 256, 0, stream>>>(x, scorer[layer], invn,
                                                           scores, d);
            size_t smem = (size_t)NNODES * 4 + 1024 * 4 + 1024 * 4;
            k_topk<<<1, 1024, smem, stream>>>(scores, tidx, tval);
            k_xtilde<<<(d * HDIM + 255) / 256, 256, 0, stream>>>(x, tidx, tval, Xt, d);
            // ---- GRU weight evolution (WMMA f32) ----
            int gblocks = ((d / 16) * (HDIM / 16) + 7) / 8;
            k_gru_gate<<<gblocks, 256, 0, stream>>>(gw[layer][0], Xt, gw[layer][1], Wq,
                                                    gw[layer][2], d, 0, ubuf,
                                                    nullptr, nullptr, nullptr);
            k_gru_gate<<<gblocks, 256, 0, stream>>>(gw[layer][3], Xt, gw[layer][4], Wq,
                                                    gw[layer][5], d, 0, rbuf,
                                                    nullptr, nullptr, nullptr);
            k_had<<<(d * HDIM + 255) / 256, 256, 0, stream>>>(rbuf, Wq, rqb, d * HDIM);
            k_gru_gate<<<gblocks, 256, 0, stream>>>(gw[layer][6], Xt, gw[layer][7], rqb,
                                                    gw[layer][8], d, 1, nullptr,
                                                    ubuf, Wq, Wn);
            { float* tmp = Wq; Wq = Wn; Wn = tmp; }
            // ---- GCN conv (layer-1 outputs for t<3 are dead code: skip) ----
            if (layer == 0 || t == T_STEPS - 1) {
                const int* esrc = edges + (size_t)t * 2 * NEDGES;
                const int* edst = esrc + NEDGES;
                k_fill<<<(NNODES + 255) / 256, 256, 0, stream>>>(so, 0.f, NNODES);
                k_fill<<<(NNODES + 255) / 256, 256, 0, stream>>>(si, 0.f, NNODES);
                k_deg<<<(NEDGES + 255) / 256, 256, 0, stream>>>(esrc, edst, so, si);
                k_degfin<<<(NNODES + 255) / 256, 256, 0, stream>>>(so, NNODES);
                k_degfin<<<(NNODES + 255) / 256, 256, 0, stream>>>(si, NNODES);
                dim3 pg(NNODES / 16, HDIM / 128);
                k_mm<<<pg, 256, 0, stream>>>(x, Wq, hbuf, NNODES, d, HDIM,
                                             so, nullptr, 0);
                k_fill<<<(NNODES * HDIM + 255) / 256, 256, 0, stream>>>(
                    agg, 0.f, NNODES * HDIM);
                long sthreads = (long)NEDGES * (HDIM / 4);
                k_scatter<<<(int)((sthreads + 255) / 256), 256, 0, stream>>>(
                    esrc, edst, hbuf, agg);
                float* dstf = featsA + (size_t)t * NNODES * HDIM;
                k_finalize<<<(NNODES * HDIM + 255) / 256, 256, 0, stream>>>(
                    agg, si, dstf, NNODES * HDIM);
            }
        }
    }
    // ---- classifier MLP ----
    dim3 mg(NNODES / 16, CHID / 128);
    k_mm<<<mg, 256, 0, stream>>>(featsA + (size_t)3 * NNODES * HDIM, mW1, hidden,
                                 NNODES, HDIM, CHID, nullptr, mb1, 1);
    k_mlp2<<<(NNODES + 7) / 8, 256, 0, stream>>>(hidden, mW2, mb2, (float*)d_out);
}